// Model_41386304864551
// MI455X (gfx1250) — compile-verified
//
#include <hip/hip_runtime.h>
#include <math.h>

// ---------------- model constants ----------------
#define B_      32
#define SEQ_    512
#define NVAR_   862
#define NMARK_  4
#define LTOK_   (NVAR_ + NMARK_)     // 866 tokens
#define DM_     512
#define DS_     16
#define DTR_    32
#define PRED_   96
#define NL_     4
#define EPS_    1e-5f
#define MROWS_  (B_ * LTOK_)         // 27712 rows for all row-major GEMMs

typedef float v2f __attribute__((ext_vector_type(2)));
typedef float v8f __attribute__((ext_vector_type(8)));

#define LDSS 20   // LDS row stride in floats: 80B -> 16B-aligned B128 stores, conflict-free

// ---- CDNA5 async global->LDS path (gfx1250), with safe fallback ----
#if defined(__has_builtin)
#  if __has_builtin(__builtin_amdgcn_global_load_async_to_lds_b128)
#    define USE_ASYNC_LDS 1
#  endif
#endif
#ifndef USE_ASYNC_LDS
#  define USE_ASYNC_LDS 0
#endif

#if USE_ASYNC_LDS
// builtin signature (probed from hipcc diagnostics):
//   arg0: generic 'int __vector(4)*'  (global source)
//   arg1: addrspace(3) 'int __vector(4)*' (LDS destination)
//   arg2/arg3: immediate offset / cpol
typedef int vi4 __attribute__((vector_size(16)));
typedef __attribute__((address_space(3))) vi4 lvi4;

__device__ __forceinline__ void async_wait_all()
{
#if defined(__has_builtin) && __has_builtin(__builtin_amdgcn_s_wait_asynccnt)
    __builtin_amdgcn_s_wait_asynccnt(0);
#else
    asm volatile("s_wait_asynccnt 0" ::: "memory");
#endif
}
#endif

// =====================================================================
// Tiled NT GEMM:  C[M,N] = A[M,K] * W[N,K]^T (+bias) (+activation)
// fp32 throughout via V_WMMA_F32_16X16X4_F32 (exact fp32 matrix math).
// Block tile 128x128, 8 waves arranged 4(M) x 2(N); wave tile 32x64.
// K panels (16-wide) are fed to LDS by GLOBAL_LOAD_ASYNC_TO_LDS_B128
// in a double-buffered pipeline (ASYNCcnt), overlapping with WMMA.
// ACT: 0 = none, 2 = exact GELU, 3 = softplus
// =====================================================================
template<int ACT>
__global__ __launch_bounds__(256)
void gemm_nt_kernel(const float* __restrict__ A, int lda,
                    const float* __restrict__ W, int ldw,
                    const float* __restrict__ bias,
                    float* __restrict__ C, int ldc,
                    int M, int N, int K)
{
#if USE_ASYNC_LDS
    __shared__ __align__(16) float As[2][128 * LDSS];
    __shared__ __align__(16) float Ws[2][128 * LDSS];
#else
    __shared__ __align__(16) float As[1][128 * LDSS];
    __shared__ __align__(16) float Ws[1][128 * LDSS];
#endif

    const int tid  = threadIdx.x;
    const int lane = tid & 31;
    const int wave = tid >> 5;
    const int wm   = wave >> 1;          // 0..3 (M)
    const int wn   = wave & 1;           // 0..1 (N)
    const int l15  = lane & 15;
    const int lh   = lane >> 4;          // lane half: selects K pair (A) / K rows 2,3 (B)
    const int mBlk = blockIdx.y * 128;
    const int nBlk = blockIdx.x * 128;

    const v8f vzero = {0.f,0.f,0.f,0.f,0.f,0.f,0.f,0.f};
    v8f acc[2][4];
#pragma unroll
    for (int i = 0; i < 2; ++i)
#pragma unroll
        for (int j = 0; j < 4; ++j) acc[i][j] = vzero;

    // per-thread tile-slot mapping: 512 float4 slots per matrix, 2 per thread
    // q -> row r = q>>2, col c = (q&3)*4
    auto compute_panel = [&](const float* Asb, const float* Wsb) {
#pragma unroll
        for (int k2 = 0; k2 < 16; k2 += 4) {
            // A fragment (16x4): lanes 0-15 hold K=k2+0/1, lanes 16-31 hold K=k2+2/3
            v2f af[2], bf[4];
#pragma unroll
            for (int i = 0; i < 2; ++i)
                af[i] = *(const v2f*)(&Asb[(wm * 32 + i * 16 + l15) * LDSS + k2 + 2 * lh]);
            // B fragment (4x16): B[k][n] = W[n][k] -> identical per-lane gather from W panel
#pragma unroll
            for (int j = 0; j < 4; ++j)
                bf[j] = *(const v2f*)(&Wsb[(wn * 64 + j * 16 + l15) * LDSS + k2 + 2 * lh]);
#pragma unroll
            for (int i = 0; i < 2; ++i)
#pragma unroll
                for (int j = 0; j < 4; ++j)
                    acc[i][j] = __builtin_amdgcn_wmma_f32_16x16x4_f32(
                        false, af[i], false, bf[j], (short)0, acc[i][j], false, false);
        }
    };

#if USE_ASYNC_LDS
    // OOB rows are clamped to a valid row: garbage lands only in C rows/cols
    // that the store guards skip, and stays finite (no NaN contamination).
    auto issue_panel = [&](float* Asb, float* Wsb, int k0) {
#pragma unroll
        for (int it = 0; it < 2; ++it) {
            int q = tid + it * 256;
            int r = q >> 2;
            int c = (q & 3) << 2;
            int ga = mBlk + r; if (ga >= M) ga = M - 1;
            __builtin_amdgcn_global_load_async_to_lds_b128(
                (vi4*)(A + (size_t)ga * lda + k0 + c),
                (lvi4*)(Asb + r * LDSS + c), 0, 0);
            int gw = nBlk + r; if (gw >= N) gw = N - 1;
            __builtin_amdgcn_global_load_async_to_lds_b128(
                (vi4*)(W + (size_t)gw * ldw + k0 + c),
                (lvi4*)(Wsb + r * LDSS + c), 0, 0);
        }
    };

    issue_panel(As[0], Ws[0], 0);                  // prologue
    int buf = 0;
    for (int k0 = 0; k0 < K; k0 += 16) {
        async_wait_all();                           // my async writes to As/Ws[buf] done
        __syncthreads();                            // everyone's writes visible
        if (k0 + 16 < K)
            issue_panel(As[buf ^ 1], Ws[buf ^ 1], k0 + 16);   // prefetch next panel
        compute_panel(As[buf], Ws[buf]);
        __syncthreads();                            // all waves done reading buf
        buf ^= 1;
    }
#else
    for (int k0 = 0; k0 < K; k0 += 16) {
#pragma unroll
        for (int it = 0; it < 2; ++it) {
            int q = tid + it * 256;
            int r = q >> 2;
            int c = (q & 3) << 2;
            float4 va = make_float4(0.f, 0.f, 0.f, 0.f);
            int ga = mBlk + r;
            if (ga < M) va = *(const float4*)(A + (size_t)ga * lda + k0 + c);
            *(float4*)(&As[0][r * LDSS + c]) = va;
            float4 vw = make_float4(0.f, 0.f, 0.f, 0.f);
            int gw = nBlk + r;
            if (gw < N) vw = *(const float4*)(W + (size_t)gw * ldw + k0 + c);
            *(float4*)(&Ws[0][r * LDSS + c]) = vw;
        }
        __syncthreads();
        compute_panel(As[0], Ws[0]);
        __syncthreads();
    }
#endif

    // epilogue: C/D layout -> VGPR r: lanes 0-15 => M=r, lanes 16-31 => M=r+8
#pragma unroll
    for (int j = 0; j < 4; ++j) {
        int col = nBlk + wn * 64 + j * 16 + l15;
        if (col >= N) continue;
        float bv = bias ? bias[col] : 0.f;
#pragma unroll
        for (int i = 0; i < 2; ++i) {
#pragma unroll
            for (int r = 0; r < 8; ++r) {
                int row = mBlk + wm * 32 + i * 16 + r + 8 * lh;
                if (row < M) {
                    float v = acc[i][j][r] + bv;
                    if (ACT == 2) {                 // exact GELU
                        v = 0.5f * v * (1.f + erff(v * 0.70710678118654752f));
                    } else if (ACT == 3) {          // softplus = logaddexp(v,0)
                        v = (v > 0.f) ? v + log1pf(__expf(-v)) : log1pf(__expf(v));
                    }
                    C[(size_t)row * ldc + col] = v;
                }
            }
        }
    }
}

// =====================================================================
// Input prep: per-(b,var) mean/std over SEQ, write normalized tokens
// transposed into tok[B,866,512]; marks appended as tokens 862..865.
// =====================================================================
__global__ __launch_bounds__(256)
void prep_kernel(const float* __restrict__ x_enc, const float* __restrict__ x_mark,
                 float* __restrict__ tok, float* __restrict__ mean_out,
                 float* __restrict__ std_out)
{
    const int n = blockIdx.x;     // token 0..865
    const int b = blockIdx.y;
    const int tid = threadIdx.x;

    if (n >= NVAR_) {
        const int m = n - NVAR_;
#pragma unroll
        for (int i = 0; i < 2; ++i) {
            int s = tid + i * 256;
            tok[((size_t)(b * LTOK_ + n)) * SEQ_ + s] =
                x_mark[((size_t)b * SEQ_ + s) * NMARK_ + m];
        }
        return;
    }

    __shared__ float ssum[256], ssq[256];
    float v0 = x_enc[((size_t)b * SEQ_ + tid) * NVAR_ + n];
    float v1 = x_enc[((size_t)b * SEQ_ + tid + 256) * NVAR_ + n];
    ssum[tid] = v0 + v1;
    ssq[tid]  = v0 * v0 + v1 * v1;
    __syncthreads();
    for (int o = 128; o > 0; o >>= 1) {
        if (tid < o) { ssum[tid] += ssum[tid + o]; ssq[tid] += ssq[tid + o]; }
        __syncthreads();
    }
    float mean = ssum[0] * (1.f / SEQ_);
    float var  = ssq[0] * (1.f / SEQ_) - mean * mean;
    float sd   = sqrtf(var + EPS_);
    if (tid == 0) { mean_out[b * NVAR_ + n] = mean; std_out[b * NVAR_ + n] = sd; }
    float inv = 1.f / sd;
    tok[((size_t)(b * LTOK_ + n)) * SEQ_ + tid]       = (v0 - mean) * inv;
    tok[((size_t)(b * LTOK_ + n)) * SEQ_ + tid + 256] = (v1 - mean) * inv;
}

// =====================================================================
// Causal depthwise conv (DCONV=2) + SiLU over token dim, reading the
// x-half of xz[.,1024]. Causality is per-batch in flattened (b,n) rows.
// =====================================================================
__global__ __launch_bounds__(256)
void conv_silu_kernel(const float* __restrict__ xz, const float* __restrict__ cw,
                      const float* __restrict__ cb, float* __restrict__ xc)
{
    size_t idx = (size_t)blockIdx.x * 256 + threadIdx.x;
    if (idx >= (size_t)MROWS_ * DM_) return;
    int d = (int)(idx & (DM_ - 1));
    size_t row = idx >> 9;
    int n = (int)(row % LTOK_);
    float x1 = xz[row * 1024 + d];
    float x0 = (n > 0) ? xz[(row - 1) * 1024 + d] : 0.f;
    float v = x0 * cw[d * 2 + 0] + x1 * cw[d * 2 + 1] + cb[d];
    xc[idx] = v / (1.f + __expf(-v));
}

// =====================================================================
// Selective scan + fused output gate: per-(b,d) state (DS=16) lives in
// registers; 866 sequential steps stream dt/x/z (coalesced) and B/C
// (wave-broadcast). Writes y = (scan + D*x) * silu(z) directly.
// =====================================================================
__global__ __launch_bounds__(256)
void scan_kernel(const float* __restrict__ dt, const float* __restrict__ dbc,
                 const float* __restrict__ x, const float* __restrict__ xz,
                 const float* __restrict__ Alog, const float* __restrict__ Dp,
                 float* __restrict__ y, int L)
{
    const int b = blockIdx.y;
    const int d = blockIdx.x * 256 + threadIdx.x;

    float A[DS_], h[DS_];
#pragma unroll
    for (int s = 0; s < DS_; ++s) {
        A[s] = -__expf(Alog[d * DS_ + s]);
        h[s] = 0.f;
    }
    const float Dd = Dp[d];
    const size_t base = (size_t)b * L;

    for (int t = 0; t < L; ++t) {
        size_t row = base + t;
        float dtv = dt[row * DM_ + d];
        float xv  = x[row * DM_ + d];
        float zv  = xz[row * 1024 + DM_ + d];
        float dx  = dtv * xv;
        const float* bc = dbc + row * 64;
        float accv = 0.f;
#pragma unroll
        for (int s = 0; s < DS_; ++s) {
            float Bv = bc[DTR_ + s];
            float Cv = bc[DTR_ + DS_ + s];
            float dA = __expf(dtv * A[s]);
            h[s] = dA * h[s] + dx * Bv;
            accv += h[s] * Cv;
        }
        float gate = zv / (1.f + __expf(-zv));
        y[row * DM_ + d] = (accv + Dd * xv) * gate;
    }
}

// token-dim flip (per batch), float4-wide copy
__global__ __launch_bounds__(256)
void flip_kernel(const float* __restrict__ src, float* __restrict__ dst)
{
    size_t idx = (size_t)blockIdx.x * 256 + threadIdx.x;
    if (idx >= (size_t)MROWS_ * (DM_ / 4)) return;
    size_t row = idx >> 7;
    int c = (int)(idx & 127);
    int bb = (int)(row / LTOK_);
    int n  = (int)(row % LTOK_);
    size_t srow = (size_t)bb * LTOK_ + (LTOK_ - 1 - n);
    ((float4*)dst)[row * 128 + c] = ((const float4*)src)[srow * 128 + c];
}

// out = LN(X [+ A1] [+ A2(flipped)]) * g + b ; one block per row, in-place safe
__global__ __launch_bounds__(256)
void resln_kernel(const float* __restrict__ X, const float* __restrict__ A1,
                  const float* __restrict__ A2, int flip,
                  const float* __restrict__ g, const float* __restrict__ bb,
                  float* __restrict__ out, int L)
{
    const int row = blockIdx.x;
    const int tid = threadIdx.x;
    __shared__ float ssum[256], ssq[256];

    int bidx = row / L, n = row % L;
    size_t frow = (size_t)bidx * L + (L - 1 - n);

    float v[2];
#pragma unroll
    for (int i = 0; i < 2; ++i) {
        int d = tid + i * 256;
        float t = X[(size_t)row * DM_ + d];
        if (A1) t += A1[(size_t)row * DM_ + d];
        if (A2) t += A2[(flip ? frow : (size_t)row) * DM_ + d];
        v[i] = t;
    }
    ssum[tid] = v[0] + v[1];
    ssq[tid]  = v[0] * v[0] + v[1] * v[1];
    __syncthreads();
    for (int o = 128; o > 0; o >>= 1) {
        if (tid < o) { ssum[tid] += ssum[tid + o]; ssq[tid] += ssq[tid + o]; }
        __syncthreads();
    }
    float mean = ssum[0] * (1.f / DM_);
    float var  = ssq[0] * (1.f / DM_) - mean * mean;
    float rstd = rsqrtf(var + EPS_);
#pragma unroll
    for (int i = 0; i < 2; ++i) {
        int d = tid + i * 256;
        out[(size_t)row * DM_ + d] = (v[i] - mean) * rstd * g[d] + bb[d];
    }
}

// head epilogue: transpose to (B,PRED,NVAR), de-normalize, append loss scalar
__global__ __launch_bounds__(256)
void head_out_kernel(const float* __restrict__ dec_tmp, const float* __restrict__ mean,
                     const float* __restrict__ sd, float* __restrict__ out)
{
    size_t idx = (size_t)blockIdx.x * 256 + threadIdx.x;
    const size_t total = (size_t)B_ * PRED_ * NVAR_;
    if (idx >= total) return;
    int v = (int)(idx % NVAR_);
    int t = (int)((idx / NVAR_) % PRED_);
    int b = (int)(idx / ((size_t)NVAR_ * PRED_));
    float val = dec_tmp[((size_t)b * LTOK_ + v) * PRED_ + t];
    out[idx] = val * sd[b * NVAR_ + v] + mean[b * NVAR_ + v];
    if (idx == 0) out[total] = 0.f;   // loss_reg
}

// ---------------- host-side orchestration ----------------
static inline int cdiv_i(int a, int b) { return (a + b - 1) / b; }

template<int ACT>
static void gemm(hipStream_t s, const float* A, int lda, const float* W, int ldw,
                 const float* bias, float* C, int ldc, int M, int N, int K)
{
    dim3 g(cdiv_i(N, 128), cdiv_i(M, 128));
    gemm_nt_kernel<ACT><<<g, dim3(256), 0, s>>>(A, lda, W, ldw, bias, C, ldc, M, N, K);
}

extern "C" void kernel_launch(void* const* d_in, const int* in_sizes, int n_in,
                              void* d_out, int out_size, void* d_ws, size_t ws_size,
                              hipStream_t stream)
{
    (void)in_sizes; (void)n_in; (void)out_size; (void)ws_size;

    const float* x_enc  = (const float*)d_in[0];
    const float* x_mark = (const float*)d_in[1];
    const float* W_emb  = (const float*)d_in[4];
    const float* b_emb  = (const float*)d_in[5];
    const float* mWin   = (const float*)d_in[6];
    const float* mcw    = (const float*)d_in[7];
    const float* mcb    = (const float*)d_in[8];
    const float* mWx    = (const float*)d_in[9];
    const float* mWdt   = (const float*)d_in[10];
    const float* mbdt   = (const float*)d_in[11];
    const float* mAlog  = (const float*)d_in[12];
    const float* mD     = (const float*)d_in[13];
    const float* mWout  = (const float*)d_in[14];
    const float* W1     = (const float*)d_in[15];
    const float* b1     = (const float*)d_in[16];
    const float* W2     = (const float*)d_in[17];
    const float* b2     = (const float*)d_in[18];
    const float* ln1g   = (const float*)d_in[19];
    const float* ln1b   = (const float*)d_in[20];
    const float* ln2g   = (const float*)d_in[21];
    const float* ln2b   = (const float*)d_in[22];
    const float* lnfg   = (const float*)d_in[23];
    const float* lnfb   = (const float*)d_in[24];
    const float* headW  = (const float*)d_in[25];
    const float* headb  = (const float*)d_in[26];

    // workspace layout (floats); total ~131M floats (~525 MB)
    float* ws = (float*)d_ws;
    const size_t S1 = (size_t)MROWS_ * DM_;
    float* h     = ws;                 // persistent hidden state
    float* tmpu  = ws + 1 * S1;        // tok buffer, then flipped-input buffer
    float* xz    = ws + 2 * S1;        // (M,1024)
    float* xc    = ws + 4 * S1;        // conv+silu output
    float* dtb   = ws + 5 * S1;        // softplus(dt)
    float* yb    = ws + 6 * S1;        // gated scan output / ffn mid
    float* fwd   = ws + 7 * S1;
    float* rev   = ws + 8 * S1;
    float* dbc   = ws + 9 * S1;                         // (M,64)
    float* dect  = dbc  + (size_t)MROWS_ * 64;          // (M,96)
    float* meanb = dect + (size_t)MROWS_ * PRED_;
    float* stdb  = meanb + (size_t)B_ * NVAR_;

    const int Mr = MROWS_;
    const int elemGrid = cdiv_i(Mr * DM_, 256);

    // 1) normalize + tokenize
    prep_kernel<<<dim3(LTOK_, B_), 256, 0, stream>>>(x_enc, x_mark, tmpu, meanb, stdb);

    // 2) embedding: h = tok · W_emb^T + b_emb   (K = SEQ = 512)
    gemm<0>(stream, tmpu, SEQ_, W_emb, SEQ_, b_emb, h, DM_, Mr, DM_, SEQ_);

    // 3) layers
    for (int l = 0; l < NL_; ++l) {
        for (int dir = 0; dir < 2; ++dir) {
            const size_t w2 = (size_t)(l * 2 + dir);
            const float* Win  = mWin  + w2 * 1024 * 512;
            const float* cw   = mcw   + w2 * 512 * 2;
            const float* cb   = mcb   + w2 * 512;
            const float* Wx   = mWx   + w2 * 64 * 512;
            const float* Wdt  = mWdt  + w2 * 512 * 32;
            const float* bdt  = mbdt  + w2 * 512;
            const float* Alog = mAlog + w2 * 512 * 16;
            const float* Dp   = mD    + w2 * 512;
            const float* Wout = mWout + w2 * 512 * 512;

            const float* u = h;
            float* outb = fwd;
            if (dir == 1) {
                flip_kernel<<<cdiv_i(Mr * (DM_ / 4), 256), 256, 0, stream>>>(h, tmpu);
                u = tmpu;
                outb = rev;
            }

            gemm<0>(stream, u, DM_, Win, DM_, nullptr, xz, 2 * DM_, Mr, 2 * DM_, DM_);
            conv_silu_kernel<<<elemGrid, 256, 0, stream>>>(xz, cw, cb, xc);
            gemm<0>(stream, xc, DM_, Wx, DM_, nullptr, dbc, 64, Mr, 64, DM_);
            gemm<3>(stream, dbc, 64, Wdt, DTR_, bdt, dtb, DM_, Mr, DM_, DTR_);   // softplus
            scan_kernel<<<dim3(DM_ / 256, B_), 256, 0, stream>>>(dtb, dbc, xc, xz,
                                                                 Alog, Dp, yb, LTOK_);
            gemm<0>(stream, yb, DM_, Wout, DM_, nullptr, outb, DM_, Mr, DM_, DM_);
        }

        // h = LN1(h + fwd + rev[flipped])
        resln_kernel<<<Mr, 256, 0, stream>>>(h, fwd, rev, 1,
                                             ln1g + (size_t)l * DM_, ln1b + (size_t)l * DM_,
                                             h, LTOK_);
        // FFN: yb = gelu(h·W1^T + b1); fwd = yb·W2^T + b2; h = LN2(h + fwd)
        gemm<2>(stream, h, DM_, W1 + (size_t)l * DM_ * DM_, DM_,
                b1 + (size_t)l * DM_, yb, DM_, Mr, DM_, DM_);
        gemm<0>(stream, yb, DM_, W2 + (size_t)l * DM_ * DM_, DM_,
                b2 + (size_t)l * DM_, fwd, DM_, Mr, DM_, DM_);
        resln_kernel<<<Mr, 256, 0, stream>>>(h, fwd, nullptr, 0,
                                             ln2g + (size_t)l * DM_, ln2b + (size_t)l * DM_,
                                             h, LTOK_);
    }

    // 4) final LN + head + de-normalized output (+ loss scalar)
    resln_kernel<<<Mr, 256, 0, stream>>>(h, nullptr, nullptr, 0, lnfg, lnfb, h, LTOK_);
    gemm<0>(stream, h, DM_, headW, DM_, headb, dect, PRED_, Mr, PRED_, DM_);
    head_out_kernel<<<cdiv_i(B_ * PRED_ * NVAR_, 256), 256, 0, stream>>>(
        dect, meanb, stdb, (float*)d_out);
}